// Attention_86423331930641
// MI455X (gfx1250) — compile-verified
//
#include <hip/hip_runtime.h>
#include <hip/hip_bf16.h>
#include <math.h>

typedef __attribute__((ext_vector_type(2))) float v2f;
typedef __attribute__((ext_vector_type(8))) float v8f;

#define N_NODES 8192
#define F_IN    256
#define F_OUT   64

__device__ __forceinline__ float leaky02(float x) {
    return x > 0.0f ? x : 0.2f * x;
}

// ---------------------------------------------------------------------------
// Kernel 1: xf[8192,64] = X[8192,256] @ W[64,256]^T   (f32 WMMA 16x16x4)
// One block per 16-row tile, 4 waves, wave w owns output cols [16w,16w+16).
// A-frag (16x4 f32): lane<16 holds K=0,1 of row=lane; lane>=16 holds K=2,3.
// B-frag (4x16 f32): mirrored — lane<16 holds K=0,1 of col=lane; etc.
// ---------------------------------------------------------------------------
__global__ __launch_bounds__(128) void gat_xw_kernel(const float* __restrict__ X,
                                                     const float* __restrict__ W,
                                                     float* __restrict__ xf) {
    const int ib    = blockIdx.x;            // 16-row tile of X
    const int lane  = threadIdx.x & 31;
    const int wave  = threadIdx.x >> 5;      // 0..3 -> f16-tile
    const int fbase = wave * 16;
    const int row   = lane & 15;
    const int koff  = (lane >> 4) << 1;      // 0 or 2
    const int grow  = ib * 16 + row;
    const int fcol  = fbase + (lane & 15);

    v8f acc = {0.f, 0.f, 0.f, 0.f, 0.f, 0.f, 0.f, 0.f};
    for (int k = 0; k < F_IN; k += 4) {
        v2f a, b;
        a.x = X[grow * F_IN + k + koff];
        a.y = X[grow * F_IN + k + koff + 1];
        // B[k][n] = (W^T)[k][n] = W[n][k]
        b.x = W[fcol * F_IN + k + koff];
        b.y = W[fcol * F_IN + k + koff + 1];
        acc = __builtin_amdgcn_wmma_f32_16x16x4_f32(false, a, false, b,
                                                    (short)0, acc, false, false);
    }
    // D layout: vgpr v -> M = v + 8*(lane>=16), N = lane&15
    const int mbase = (lane >> 4) << 3;
#pragma unroll
    for (int v = 0; v < 8; ++v) {
        xf[(ib * 16 + mbase + v) * F_OUT + fbase + (lane & 15)] = acc[v];
    }
}

// ---------------------------------------------------------------------------
// Kernel 2: s[i] = xf[i,:] . a[0], t[i] = xf[i,:] . a[1]
// ---------------------------------------------------------------------------
__global__ __launch_bounds__(256) void gat_st_kernel(const float* __restrict__ xf,
                                                     const float* __restrict__ avec,
                                                     float* __restrict__ s,
                                                     float* __restrict__ t) {
    int i = blockIdx.x * blockDim.x + threadIdx.x;
    if (i >= N_NODES) return;
    float s0 = 0.0f, s1 = 0.0f;
#pragma unroll 8
    for (int f = 0; f < F_OUT; ++f) {
        float v = xf[i * F_OUT + f];
        s0 += v * avec[f];
        s1 += v * avec[F_OUT + f];
    }
    s[i] = s0;
    t[i] = s1;
}

// ---------------------------------------------------------------------------
// Kernel 3: zero colsum
// ---------------------------------------------------------------------------
__global__ void gat_zero_kernel(float* __restrict__ colsum) {
    int i = blockIdx.x * blockDim.x + threadIdx.x;
    if (i < N_NODES) colsum[i] = 0.0f;
}

// ---------------------------------------------------------------------------
// Kernel 4: colsum[j] = sum_i A[i,j] * exp(leaky(s[i] + t[j]))
// grid (32 col-blocks, 64 row-blocks), 256 threads; thread -> one column,
// 128 rows each; coalesced row-major sweeps of A; atomicAdd per column.
// ---------------------------------------------------------------------------
__global__ __launch_bounds__(256) void gat_colsum_kernel(const float* __restrict__ A,
                                                         const float* __restrict__ s,
                                                         const float* __restrict__ t,
                                                         float* __restrict__ colsum) {
    const int j  = blockIdx.x * 256 + threadIdx.x;
    const int i0 = blockIdx.y * 128;
    const float tv = t[j];
    float acc = 0.0f;
    for (int r = 0; r < 128; ++r) {
        const int i = i0 + r;
        float av = A[(size_t)i * N_NODES + j];
        if (av > 0.0f) {
            acc += __expf(leaky02(s[i] + tv));
        }
    }
    atomicAdd(&colsum[j], acc);
}

// ---------------------------------------------------------------------------
// Kernel 5: q[j,f] = xf[j,f] / colsum[j]
// ---------------------------------------------------------------------------
__global__ __launch_bounds__(256) void gat_q_kernel(const float* __restrict__ xf,
                                                    const float* __restrict__ colsum,
                                                    float* __restrict__ q) {
    int idx = blockIdx.x * blockDim.x + threadIdx.x;
    if (idx >= N_NODES * F_OUT) return;
    int j = idx / F_OUT;
    q[idx] = xf[idx] / colsum[j];
}

// ---------------------------------------------------------------------------
// Kernel 6: out[i,f] = sum_j p[i,j] * q[j,f]   (f32 WMMA 16x16x4)
// One block per 16-row tile; P(16 x 128) chunks staged in LDS (exp fused);
// 4 waves each own a 16-col slice of F1. q (2MB) is L2-resident.
// ---------------------------------------------------------------------------
#define CHUNK 128

__global__ __launch_bounds__(128) void gat_aggregate_kernel(const float* __restrict__ A,
                                                            const float* __restrict__ s,
                                                            const float* __restrict__ t,
                                                            const float* __restrict__ q,
                                                            float* __restrict__ out) {
    __shared__ float pt[16 * CHUNK];   // 8 KB

    const int ib    = blockIdx.x;            // 16-row output tile
    const int tid   = threadIdx.x;           // 0..127
    const int lane  = tid & 31;
    const int wave  = tid >> 5;
    const int fbase = wave * 16;
    const int row   = lane & 15;
    const int koff  = (lane >> 4) << 1;      // 0 or 2
    const int fcol  = fbase + (lane & 15);

    v8f acc = {0.f, 0.f, 0.f, 0.f, 0.f, 0.f, 0.f, 0.f};

    for (int ch = 0; ch < N_NODES / CHUNK; ++ch) {
        const int j0 = ch * CHUNK;
        // ---- stage P tile: pt[r][c] = A[ib*16+r, j0+c] * exp(leaky(s+t)) ----
        const float tv = t[j0 + tid];
#pragma unroll 4
        for (int r = 0; r < 16; ++r) {
            const int gi = ib * 16 + r;
            float av = A[(size_t)gi * N_NODES + j0 + tid];
            float pv = 0.0f;
            if (av > 0.0f) pv = __expf(leaky02(s[gi] + tv));
            pt[r * CHUNK + tid] = pv;
        }
        __syncthreads();

        // ---- 32 WMMA k-steps over this chunk ----
        for (int kk = 0; kk < CHUNK / 4; ++kk) {
            v2f a, b;
            a.x = pt[row * CHUNK + kk * 4 + koff];
            a.y = pt[row * CHUNK + kk * 4 + koff + 1];
            const int jj = j0 + kk * 4 + koff;
            b.x = q[jj * F_OUT + fcol];
            b.y = q[(jj + 1) * F_OUT + fcol];
            acc = __builtin_amdgcn_wmma_f32_16x16x4_f32(false, a, false, b,
                                                        (short)0, acc, false, false);
        }
        __syncthreads();   // protect pt before next staging
    }

    const int mbase = (lane >> 4) << 3;
#pragma unroll
    for (int v = 0; v < 8; ++v) {
        out[(ib * 16 + mbase + v) * F_OUT + fbase + (lane & 15)] = acc[v];
    }
}

// ---------------------------------------------------------------------------
// Host launcher
// ---------------------------------------------------------------------------
extern "C" void kernel_launch(void* const* d_in, const int* in_sizes, int n_in,
                              void* d_out, int out_size, void* d_ws, size_t ws_size,
                              hipStream_t stream) {
    const float* X    = (const float*)d_in[0];   // [8192,256]
    const float* A    = (const float*)d_in[1];   // [8192,8192]
    const float* W    = (const float*)d_in[2];   // [64,256]
    const float* avec = (const float*)d_in[3];   // [2,64,1] flat

    float* ws     = (float*)d_ws;
    float* xf     = ws;                               // 8192*64
    float* s      = xf + N_NODES * F_OUT;             // 8192
    float* t      = s + N_NODES;                      // 8192
    float* colsum = t + N_NODES;                      // 8192
    float* q      = colsum + N_NODES;                 // 8192*64

    float* out = (float*)d_out;                       // [8192,64]

    // 1) xf = X @ W^T
    gat_xw_kernel<<<N_NODES / 16, 128, 0, stream>>>(X, W, xf);
    // 2) s, t
    gat_st_kernel<<<N_NODES / 256, 256, 0, stream>>>(xf, avec, s, t);
    // 3) zero colsum
    gat_zero_kernel<<<N_NODES / 256, 256, 0, stream>>>(colsum);
    // 4) column sums of softmax numerators
    dim3 gcs(N_NODES / 256, N_NODES / 128);
    gat_colsum_kernel<<<gcs, 256, 0, stream>>>(A, s, t, colsum);
    // 5) q = xf / colsum
    gat_q_kernel<<<(N_NODES * F_OUT) / 256, 256, 0, stream>>>(xf, colsum, q);
    // 6) out = P @ q
    gat_aggregate_kernel<<<N_NODES / 16, 128, 0, stream>>>(A, s, t, q, out);
}